// simpleLSTM_76501957476588
// MI455X (gfx1250) — compile-verified
//
#include <hip/hip_runtime.h>
#include <hip/hip_bf16.h>

typedef __attribute__((ext_vector_type(2))) float v2f;
typedef __attribute__((ext_vector_type(8))) float v8f;

// ---------------------------------------------------------------------------
// Kernel 1: layer-0 input projection for ALL timesteps via fp32 WMMA.
//   preT[tile][g][s] = x[t0+s] . w_ih0[g]*sc + (b_ih0[g]+b_hh0[g])*sc
// where sc=2 for the tanh (g) gates 4,5 (folds tanh(x)=2*sigmoid(2x)-1 prescale)
// Gate-major tile layout so the scan kernel needs NO cross-lane shuffles.
// ---------------------------------------------------------------------------
__global__ __launch_bounds__(256) void lstm_pregates_wmma(
    const float* __restrict__ x,       // (T,2)
    const float* __restrict__ w_ih0,   // (8,2)
    const float* __restrict__ b_ih0,   // (8,)
    const float* __restrict__ b_hh0,   // (8,)
    float* __restrict__ preT,          // (T/16, 8, 16) gate-major tiles
    int T)
{
    const int lane = threadIdx.x & 31;
    const int wave = threadIdx.x >> 5;
    const int tile = blockIdx.x * (blockDim.x >> 5) + wave;
    const int t0   = tile * 16;
    if (t0 >= T) return;

    const int  m  = lane & 15;     // M (timestep-in-tile) for A; N (gate) for B/C/D
    const bool hi = lane >= 16;    // half-wave: K=2,3 for A/B (zero pad); M+=8 for C/D

    // A-matrix 16x4 fp32: lanes 0-15 VGPR0=K0, VGPR1=K1; lanes 16-31 K2,K3 (zeros)
    v2f A = {0.f, 0.f};
    if (!hi) {
        A.x = x[(t0 + m) * 2 + 0];
        A.y = x[(t0 + m) * 2 + 1];
    }
    // B-matrix 4x16 fp32 (B[k][n] = w_ih0[n][k] * sc); rows K=2,3 / cols n>=8 zero
    const float sc = (m == 4 || m == 5) ? 2.f : 1.f;
    v2f B = {0.f, 0.f};
    if (!hi && m < 8) {
        B.x = w_ih0[m * 2 + 0] * sc;
        B.y = w_ih0[m * 2 + 1] * sc;
    }
    // C-matrix: (pre-scaled) bias broadcast down each column n
    const float bias = (m < 8) ? (b_ih0[m] + b_hh0[m]) * sc : 0.f;
    v8f C;
#pragma unroll
    for (int v = 0; v < 8; ++v) C[v] = bias;

    v8f D = __builtin_amdgcn_wmma_f32_16x16x4_f32(
        /*neg_a=*/false, A, /*neg_b=*/false, B,
        /*c_mod=*/(short)0, C, /*reuse_a=*/false, /*reuse_b=*/false);

    // D: VGPR v = step (v + hi*8) of gate m. Store gate-major: two b128 per lane.
    if (m < 8) {
        float4 lo = {D[0], D[1], D[2], D[3]};
        float4 hi4 = {D[4], D[5], D[6], D[7]};
        float4* dst = (float4*)(preT + tile * 128 + m * 16 + (hi ? 8 : 0));
        dst[0] = lo;
        dst[1] = hi4;
    }
}

// ---------------------------------------------------------------------------
// Kernel 2: serial recurrence, one wave, zero LDS traffic in the main loop.
// Lanes 0-7 own gates (torch order i0 i1 f0 f1 g0 g1 o0 o1); broadcasts of the
// activated gates use v_readlane (VALU->SGPR), not ds_bpermute.
// ---------------------------------------------------------------------------
__device__ __forceinline__ float rdl(float v, int l) {
    return __int_as_float(__builtin_amdgcn_readlane(__float_as_int(v), l));
}
// sigmoid(x) = rcp(1 + 2^(x * -log2(e)))
#define KSIG (-1.4426950408889634f)
// sigmoid(2c) for tanh(c) = 2*sigmoid(2c)-1
#define KTAN (-2.8853900817779268f)
__device__ __forceinline__ float sig2(float x, float k) {
    return __builtin_amdgcn_rcpf(1.f + __builtin_amdgcn_exp2f(x * k));
}

__global__ __launch_bounds__(32) void lstm_scan(
    const float* __restrict__ preT,    // (T/16, 8, 16) from kernel 1 (g pre-scaled)
    const float* __restrict__ hprev,   // [h00,h01,h10,h11]
    const float* __restrict__ cprev,
    const float* __restrict__ w_hh0,   // (8,2)
    const float* __restrict__ w_ih1,   // (8,2)
    const float* __restrict__ w_hh1,   // (8,2)
    const float* __restrict__ b_ih1,   // (8,)
    const float* __restrict__ b_hh1,   // (8,)
    const float* __restrict__ fc1_w,   // (16,32)
    const float* __restrict__ fc1_b,   // (16,)
    const float* __restrict__ fc2_w,   // (2,16)
    const float* __restrict__ fc2_b,   // (2,)
    float* __restrict__ out,           // [out(2), hn(4), cn(4)]
    int nTiles)
{
    const int lane = threadIdx.x;
    const int j    = lane & 7;
    const bool isg = (j == 4) || (j == 5);
    const float gsc  = isg ? 2.f : 1.f;   // pre-scale for tanh gates
    const float amul = isg ? 2.f : 1.f;   // act = amul*sigmoid + aadd
    const float aadd = isg ? -1.f : 0.f;

    // per-gate weights, g-rows pre-scaled (selects hoisted out of the loop)
    const float wh0a = w_hh0[j * 2 + 0] * gsc, wh0b = w_hh0[j * 2 + 1] * gsc;
    const float wi1a = w_ih1[j * 2 + 0] * gsc, wi1b = w_ih1[j * 2 + 1] * gsc;
    const float wh1a = w_hh1[j * 2 + 0] * gsc, wh1b = w_hh1[j * 2 + 1] * gsc;
    const float bias1 = (b_ih1[j] + b_hh1[j]) * gsc;

    float h00 = hprev[0], h01 = hprev[1], h10 = hprev[2], h11 = hprev[3];
    float c00 = cprev[0], c01 = cprev[1], c10 = cprev[2], c11 = cprev[3];

    __shared__ float tail[32];
    __shared__ float r16[16];

    // lane j holds its own gate's 16 pre-gates for the tile: 4 x b128, no shuffles
    const float4* q0 = (const float4*)(preT + j * 16);
    float4 A0 = q0[0], A1 = q0[1], A2 = q0[2], A3 = q0[3];

#define PRE(tt) ((tt) < 4  ? ((tt)==0 ? A0.x : (tt)==1 ? A0.y : (tt)==2  ? A0.z : A0.w) \
               : (tt) < 8  ? ((tt)==4 ? A1.x : (tt)==5 ? A1.y : (tt)==6  ? A1.z : A1.w) \
               : (tt) < 12 ? ((tt)==8 ? A2.x : (tt)==9 ? A2.y : (tt)==10 ? A2.z : A2.w) \
                           : ((tt)==12? A3.x : (tt)==13? A3.y : (tt)==14 ? A3.z : A3.w))

    for (int tile = 0; tile < nTiles; ++tile) {
        float4 N0 = {0,0,0,0}, N1 = {0,0,0,0}, N2 = {0,0,0,0}, N3 = {0,0,0,0};
        if (tile + 1 < nTiles) {  // one-tile-ahead prefetch hides L2 latency
            const float4* q = (const float4*)(preT + (tile + 1) * 128 + j * 16);
            N0 = q[0]; N1 = q[1]; N2 = q[2]; N3 = q[3];
        }
        const bool lastTile = (tile == nTiles - 1);

#pragma unroll
        for (int t = 0; t < 16; ++t) {
            const float pg = PRE(t);

            // layer-1 recurrent partial from PREVIOUS h1 (off the critical chain)
            const float p2 = fmaf(wh1a, h10, wh1b * h11);

            // ---- layer 0 ----
            const float x0 = fmaf(wh0b, h01, fmaf(wh0a, h00, pg));
            const float s0 = sig2(x0, KSIG);
            const float a0 = fmaf(amul, s0, aadd);

            const float i0 = rdl(a0, 0), i1 = rdl(a0, 1);
            const float f0 = rdl(a0, 2), f1 = rdl(a0, 3);
            const float g0 = rdl(a0, 4), g1 = rdl(a0, 5);
            const float o0 = rdl(a0, 6), o1 = rdl(a0, 7);
            c00 = fmaf(f0, c00, i0 * g0);
            c01 = fmaf(f1, c01, i1 * g1);
            const float u0 = sig2(c00, KTAN);     // sigmoid(2c)
            const float u1 = sig2(c01, KTAN);
            h00 = fmaf(o0 + o0, u0, -o0);          // o * tanh(c)
            h01 = fmaf(o1 + o1, u1, -o1);

            // ---- layer 1 ----
            const float x1 = fmaf(wi1b, h01, fmaf(wi1a, h00, bias1)) + p2;
            const float s1 = sig2(x1, KSIG);
            const float a1 = fmaf(amul, s1, aadd);

            const float I0 = rdl(a1, 0), I1 = rdl(a1, 1);
            const float F0 = rdl(a1, 2), F1 = rdl(a1, 3);
            const float G0 = rdl(a1, 4), G1 = rdl(a1, 5);
            const float O0 = rdl(a1, 6), O1 = rdl(a1, 7);
            c10 = fmaf(F0, c10, I0 * G0);
            c11 = fmaf(F1, c11, I1 * G1);
            const float v0 = sig2(c10, KTAN);
            const float v1 = sig2(c11, KTAN);
            h10 = fmaf(O0 + O0, v0, -O0);
            h11 = fmaf(O1 + O1, v1, -O1);

            if (lastTile && lane == 0) {          // final tile == last 16 steps
                tail[2 * t + 0] = h10;
                tail[2 * t + 1] = h11;
            }
        }
        A0 = N0; A1 = N1; A2 = N2; A3 = N3;
    }
#undef PRE

    __syncthreads();

    // FC head: Linear(32,16) -> ReLU -> Linear(16,2)
    if (lane < 16) {
        float acc = fc1_b[lane];
#pragma unroll
        for (int k = 0; k < 32; ++k) acc = fmaf(tail[k], fc1_w[lane * 32 + k], acc);
        r16[lane] = fmaxf(acc, 0.f);
    }
    __syncthreads();
    if (lane < 2) {
        float acc = fc2_b[lane];
#pragma unroll
        for (int k = 0; k < 16; ++k) acc = fmaf(r16[k], fc2_w[lane * 16 + k], acc);
        out[lane] = acc;
    }
    if (lane == 0) {
        out[2] = h00; out[3] = h01; out[4] = h10; out[5] = h11;  // hn
        out[6] = c00; out[7] = c01; out[8] = c10; out[9] = c11;  // cn
    }
}

// ---------------------------------------------------------------------------
extern "C" void kernel_launch(void* const* d_in, const int* in_sizes, int n_in,
                              void* d_out, int out_size, void* d_ws, size_t ws_size,
                              hipStream_t stream) {
    const float* input = (const float*)d_in[0];
    const float* hprev = (const float*)d_in[1];
    const float* cprev = (const float*)d_in[2];
    const float* w_ih0 = (const float*)d_in[3];
    const float* w_hh0 = (const float*)d_in[4];
    const float* b_ih0 = (const float*)d_in[5];
    const float* b_hh0 = (const float*)d_in[6];
    const float* w_ih1 = (const float*)d_in[7];
    const float* w_hh1 = (const float*)d_in[8];
    const float* b_ih1 = (const float*)d_in[9];
    const float* b_hh1 = (const float*)d_in[10];
    const float* fc1_w = (const float*)d_in[11];
    const float* fc1_b = (const float*)d_in[12];
    const float* fc2_w = (const float*)d_in[13];
    const float* fc2_b = (const float*)d_in[14];
    float* out  = (float*)d_out;
    float* preT = (float*)d_ws;                 // (T/16,8,16) fp32 = 512 KB, L2-resident

    const int T = in_sizes[0] / 2;              // (T,1,2) flat
    const int tiles = T / 16;                   // 1024
    const int wavesPerBlock = 8;                // 256 threads
    dim3 block(32 * wavesPerBlock);
    dim3 grid((tiles + wavesPerBlock - 1) / wavesPerBlock);

    lstm_pregates_wmma<<<grid, block, 0, stream>>>(input, w_ih0, b_ih0, b_hh0, preT, T);
    lstm_scan<<<1, 32, 0, stream>>>(preT, hprev, cprev, w_hh0, w_ih1, w_hh1,
                                    b_ih1, b_hh1, fc1_w, fc1_b, fc2_w, fc2_b,
                                    out, tiles);
}